// Phrase_Generate_32916629357173
// MI455X (gfx1250) — compile-verified
//
#include <hip/hip_runtime.h>
#include <hip/hip_bf16.h>
#include <math.h>

// ---------------- problem constants ----------------
#define BATCH 128
#define LTXT  257
#define LW    256     // L-1 word positions
#define TV    1024    // video frames
#define CDIM  256
#define NH    8
#define DH    32      // CDIM/NH
#define NL    3
#define NP    8       // NUM_PHRASE
#define MDIST 2

#define NEG_INF (-__builtin_inff())

typedef float v2f __attribute__((ext_vector_type(2)));
typedef float v8f __attribute__((ext_vector_type(8)));

// GEMM tiling
#define BK 64
#define LDSS 68       // 64 + 4 pad: keeps float4 rows 16B aligned, kills bank conflicts

// ---------------- workspace layout (float offsets) ----------------
constexpr size_t WPE_OFF = 0;                       // (B,LW,C)  8388608
constexpr size_t WP_OFF  = 8388608;                 // wp, later reused as KP
constexpr size_t VPV_OFF = 16777216;                // vp(video) 33554432, later reused as VP2
constexpr size_t CUM_OFF = 50331648;                // (B,LW) cumsum
constexpr size_t ENT_OFF = CUM_OFF + (size_t)BATCH*LW;       // entropy
constexpr size_t SEL_OFF = ENT_OFF + (size_t)BATCH*LW;       // (B,NP) ints
constexpr size_t X_OFF   = SEL_OFF + (size_t)BATCH*NP;       // phrase slots (B,NP,C)
constexpr size_t QP_OFF  = X_OFF   + (size_t)BATCH*NP*CDIM;
constexpr size_t CTX_OFF = QP_OFF  + (size_t)BATCH*NP*CDIM;
constexpr size_t AO_OFF  = CTX_OFF + (size_t)BATCH*NP*CDIM;  // attn_out / FFN update
constexpr size_t AH_OFF  = AO_OFF  + (size_t)BATCH*NP*CDIM;  // per-head attn (B,NH,NP,LW)
constexpr size_t WQF_OFF = AH_OFF  + (size_t)BATCH*NH*NP*LW; // folded weights
constexpr size_t WKF_OFF = WQF_OFF + (size_t)CDIM*CDIM;
constexpr size_t WVF_OFF = WKF_OFF + (size_t)CDIM*CDIM;
constexpr size_t BQF_OFF = WVF_OFF + (size_t)CDIM*CDIM;
constexpr size_t BKF_OFF = BQF_OFF + CDIM;
constexpr size_t BVF_OFF = BKF_OFF + CDIM;

// ---------------- output layout (float offsets) ----------------
constexpr size_t OUT_PS  = 0;                                   // phrase_slot (B,NP,C)
constexpr size_t OUT_AW  = (size_t)BATCH*NP*CDIM;               // attn_weights (B,LW,TV)
constexpr size_t OUT_SS  = OUT_AW + (size_t)BATCH*LW*TV;        // slot_sim (B,NP,LW)
constexpr size_t OUT_EOS = OUT_SS + (size_t)BATCH*NP*LW;        // eos (B,1,C)

// =======================================================================
// inclusive prefix-sum of word mask per batch row
// =======================================================================
__global__ __launch_bounds__(LW)
void k_scan_mask(const unsigned char* __restrict__ tmask, float* __restrict__ cum) {
    __shared__ float s[LW];
    int b = blockIdx.x, t = threadIdx.x;
    s[t] = (tmask[(size_t)b * LTXT + 1 + t] != 0) ? 1.0f : 0.0f;
    __syncthreads();
    for (int off = 1; off < LW; off <<= 1) {
        float v = (t >= off) ? s[t - off] : 0.0f;
        __syncthreads();
        s[t] += v;
        __syncthreads();
    }
    cum[(size_t)b * LW + t] = s[t];
}

// =======================================================================
// word_pe = txt_emb[:,1:] + sine positional encoding
// =======================================================================
__global__ __launch_bounds__(CDIM)
void k_build_wpe(const float* __restrict__ txt, const float* __restrict__ cum,
                 float* __restrict__ wpe) {
    int row = blockIdx.x;            // b*LW + l
    int b = row >> 8, l = row & 255;
    int c = threadIdx.x;
    float x = cum[row];
    float ex = (float)(c & ~1) / (float)CDIM;
    float dt = powf(10000.0f, ex);
    float v = x / dt;
    float pe = (c & 1) ? cosf(v) : sinf(v);
    wpe[(size_t)row * CDIM + c] = txt[((size_t)b * LTXT + 1 + l) * CDIM + c] + pe;
}

// =======================================================================
// LDS-tiled fp32 WMMA GEMM: C = A(MxK) @ B(KxN) (+bias) (+relu), row-major.
// Block = 256 thr = 8 waves; block tile 64x64; K chunked by 64 through LDS.
// Wave (wm,wn) owns two stacked 16x16 C tiles sharing one B fragment.
// Requires M%64==0, N%64==0, K%64==0 (true for every GEMM in this net).
// =======================================================================
__device__ __forceinline__ v8f wmma4(v2f a, v2f b, v8f c) {
    return __builtin_amdgcn_wmma_f32_16x16x4_f32(false, a, false, b, (short)0, c,
                                                 false, false);
}

__global__ __launch_bounds__(256)
void k_gemm64(const float* __restrict__ A, int lda,
              const float* __restrict__ Bm, int ldb,
              const float* __restrict__ bias,
              float* __restrict__ Cm, int ldc,
              int n64, int K, int relu) {
    __shared__ float sA[64 * LDSS];
    __shared__ float sB[64 * LDSS];
    int blk = blockIdx.x;
    int m0 = (blk / n64) << 6;
    int n0 = (blk % n64) << 6;
    int t = threadIdx.x;
    int wave = t >> 5, lane = t & 31;
    int half = lane >> 4, l16 = lane & 15;
    int wn = wave & 3, wm = wave >> 2;             // wn: 0..3 (N tile), wm: 0..1
    int ncol = (wn << 4) + l16;                    // column within block tile
    v8f acc0 = {}, acc1 = {};
    const float* pa0 = sA + (wm * 32 + l16) * LDSS + (half << 1);
    const float* pa1 = pa0 + 16 * LDSS;
    const float* pb  = sB + (half << 1) * LDSS + ncol;

    for (int kc = 0; kc < K; kc += BK) {
        __syncthreads();
#pragma unroll
        for (int i = 0; i < 4; i++) {              // stage 64x64 A and B panels
            int linear = (i << 8) + t;             // 0..1023
            int row = linear >> 4;                 // 0..63
            int c4  = (linear & 15) << 2;          // 0..60 step 4
            float4 va = *(const float4*)(A + (size_t)(m0 + row) * lda + kc + c4);
            *(float4*)(sA + row * LDSS + c4) = va;
            float4 vb = *(const float4*)(Bm + (size_t)(kc + row) * ldb + n0 + c4);
            *(float4*)(sB + row * LDSS + c4) = vb;
        }
        __syncthreads();
        if (kc + BK < K) {                         // global_prefetch_b8 next panels
            int row = t >> 4, c4 = (t & 15) << 2;
            __builtin_prefetch(A + (size_t)(m0 + row) * lda + kc + BK + c4, 0, 0);
            __builtin_prefetch(Bm + (size_t)(kc + BK + row) * ldb + n0 + c4, 0, 0);
        }
#pragma unroll
        for (int k0 = 0; k0 < BK; k0 += 4) {
            v2f a0 = *(const v2f*)(pa0 + k0);      // A[M][k0+2h .. +1]
            v2f a1 = *(const v2f*)(pa1 + k0);
            v2f bv;
            bv.x = pb[k0 * LDSS];                  // B[k0+2h][n]
            bv.y = pb[(k0 + 1) * LDSS];
            acc0 = wmma4(a0, bv, acc0);
            acc1 = wmma4(a1, bv, acc1);
        }
    }
    float bb = bias ? bias[n0 + ncol] : 0.0f;
    int crow = m0 + wm * 32 + (half << 3);
    float* p0 = Cm + (size_t)crow * ldc + n0 + ncol;
    float* p1 = p0 + (size_t)16 * ldc;
#pragma unroll
    for (int r = 0; r < 8; r++) {
        float v0 = acc0[r] + bb, v1 = acc1[r] + bb;
        if (relu) { v0 = fmaxf(v0, 0.0f); v1 = fmaxf(v1, 0.0f); }
        p0[(size_t)r * ldc] = v0;
        p1[(size_t)r * ldc] = v1;
    }
}

// =======================================================================
// Batched sim GEMM: sim[b] = wp[b] (LWxC) @ vp[b]^T.  WMMA's B layout wants
// B[k][n] = V[n][k], which is a plain row-major read of V -> stage V rows
// exactly like A (no transpose) and read v2f fragments per lane.
// grid = BATCH * (LW/64) * (TV/64) blocks
// =======================================================================
__global__ __launch_bounds__(256)
void k_gemm64_sim(const float* __restrict__ WPb, const float* __restrict__ Vb,
                  float* __restrict__ Sb) {
    __shared__ float sA[64 * LDSS];
    __shared__ float sV[64 * LDSS];
    int blk = blockIdx.x;
    int b   = blk >> 6;                 // /64 tiles per batch
    int rem = blk & 63;
    int m0 = (rem >> 4) << 6;           // LW tile (4 groups)
    int n0 = (rem & 15) << 6;           // TV tile (16 groups)
    const float* A = WPb + (size_t)b * LW * CDIM;
    const float* V = Vb  + (size_t)b * TV * CDIM;
    float* Cm = Sb + (size_t)b * LW * TV;
    int t = threadIdx.x;
    int wave = t >> 5, lane = t & 31;
    int half = lane >> 4, l16 = lane & 15;
    int wn = wave & 3, wm = wave >> 2;
    int ncol = (wn << 4) + l16;
    v8f acc0 = {}, acc1 = {};
    const float* pa0 = sA + (wm * 32 + l16) * LDSS + (half << 1);
    const float* pa1 = pa0 + 16 * LDSS;
    const float* pv  = sV + ncol * LDSS + (half << 1);   // B[k][n] = V-row(ncol)[k]

    for (int kc = 0; kc < CDIM; kc += BK) {
        __syncthreads();
#pragma unroll
        for (int i = 0; i < 4; i++) {
            int linear = (i << 8) + t;
            int row = linear >> 4;
            int c4  = (linear & 15) << 2;
            float4 va = *(const float4*)(A + (size_t)(m0 + row) * CDIM + kc + c4);
            *(float4*)(sA + row * LDSS + c4) = va;
            float4 vv = *(const float4*)(V + (size_t)(n0 + row) * CDIM + kc + c4);
            *(float4*)(sV + row * LDSS + c4) = vv;
        }
        __syncthreads();
        if (kc + BK < CDIM) {
            int row = t >> 4, c4 = (t & 15) << 2;
            __builtin_prefetch(A + (size_t)(m0 + row) * CDIM + kc + BK + c4, 0, 0);
            __builtin_prefetch(V + (size_t)(n0 + row) * CDIM + kc + BK + c4, 0, 0);
        }
#pragma unroll
        for (int k0 = 0; k0 < BK; k0 += 4) {
            v2f a0 = *(const v2f*)(pa0 + k0);
            v2f a1 = *(const v2f*)(pa1 + k0);
            v2f bv = *(const v2f*)(pv + k0);
            acc0 = wmma4(a0, bv, acc0);
            acc1 = wmma4(a1, bv, acc1);
        }
    }
    int crow = m0 + wm * 32 + (half << 3);
    float* p0 = Cm + (size_t)crow * TV + n0 + ncol;
    float* p1 = p0 + (size_t)16 * TV;
#pragma unroll
    for (int r = 0; r < 8; r++) {
        p0[(size_t)r * TV] = acc0[r];
        p1[(size_t)r * TV] = acc1[r];
    }
}

// =======================================================================
// In-place masked softmax over TV + entropy. grid = B*LW, 256 threads x4 elems
// =======================================================================
__global__ __launch_bounds__(256)
void k_softmax_entropy(float* __restrict__ sim, const unsigned char* __restrict__ vmask,
                       float* __restrict__ ent) {
    __shared__ float red[256];
    int row = blockIdx.x;            // b*LW + l
    int b = row >> 8;
    float* p = sim + (size_t)row * TV;
    const unsigned char* mk = vmask + (size_t)b * TV;
    int t = threadIdx.x;
    float vals[4];
    float mx = NEG_INF;
#pragma unroll
    for (int i = 0; i < 4; i++) {
        int k = t + (i << 8);
        float v = (mk[k] != 0) ? p[k] : NEG_INF;
        vals[i] = v;
        mx = fmaxf(mx, v);
    }
    red[t] = mx; __syncthreads();
    for (int off = 128; off > 0; off >>= 1) {
        if (t < off) red[t] = fmaxf(red[t], red[t + off]);
        __syncthreads();
    }
    mx = red[0]; __syncthreads();
    float sum = 0.0f;
#pragma unroll
    for (int i = 0; i < 4; i++) {
        float e = (vals[i] == NEG_INF) ? 0.0f : expf(vals[i] - mx);
        vals[i] = e; sum += e;
    }
    red[t] = sum; __syncthreads();
    for (int off = 128; off > 0; off >>= 1) {
        if (t < off) red[t] += red[t + off];
        __syncthreads();
    }
    float inv = 1.0f / red[0]; __syncthreads();
    float epart = 0.0f;
#pragma unroll
    for (int i = 0; i < 4; i++) {
        int k = t + (i << 8);
        float a = vals[i] * inv;
        p[k] = a;
        epart -= a * logf(a + 1e-6f);
    }
    red[t] = epart; __syncthreads();
    for (int off = 128; off > 0; off >>= 1) {
        if (t < off) red[t] += red[t + off];
        __syncthreads();
    }
    if (t == 0) ent[row] = red[0];
}

// =======================================================================
// Greedy selection (replicates argsort(-score) stable order + scan),
// parallel argmax with lower-index tie-break, early exit once NP selected.
// =======================================================================
__global__ __launch_bounds__(256)
void k_select(const float* __restrict__ ent, const unsigned char* __restrict__ tmask,
              int* __restrict__ sel) {
    __shared__ float sv[256];
    __shared__ int   si[256];
    __shared__ unsigned char vis[256];
    __shared__ int ssel[NP];
    __shared__ int scnt, sdone;
    int b = blockIdx.x, t = threadIdx.x;
    float e = (tmask[(size_t)b * LTXT + 1 + t] != 0) ? ent[(size_t)b * LW + t] : NEG_INF;
    vis[t] = 0;
    if (t < NP) ssel[t] = -1;
    if (t == 0) { scnt = 0; sdone = 0; }
    __syncthreads();
    for (int it = 0; it < LW; ++it) {
        sv[t] = vis[t] ? NEG_INF : e;
        si[t] = t;
        __syncthreads();
        for (int off = 128; off > 0; off >>= 1) {
            if (t < off) {
                float v2 = sv[t + off]; int i2 = si[t + off];
                if (v2 > sv[t] || (v2 == sv[t] && i2 < si[t])) { sv[t] = v2; si[t] = i2; }
            }
            __syncthreads();
        }
        if (t == 0) {
            float best = sv[0]; int bi = si[0];
            if (best == NEG_INF) sdone = 1;     // only masked left: never selectable
            else {
                vis[bi] = 1;
                int mind = 1000000;
                for (int j = 0; j < NP; j++)
                    if (ssel[j] >= 0) {
                        int d = bi > ssel[j] ? bi - ssel[j] : ssel[j] - bi;
                        if (d < mind) mind = d;
                    }
                if (scnt < NP && mind >= MDIST) { ssel[scnt] = bi; scnt++; }
                if (scnt == NP) sdone = 1;       // later order entries cannot change sel
            }
        }
        __syncthreads();
        if (sdone) break;
    }
    if (t < NP) {
        int cnt = scnt;
        int last = ssel[(cnt > 0) ? (cnt - 1) : 0];
        sel[b * NP + t] = (t < cnt) ? ssel[t] : last;
    }
}

// gather phrase slots from word_pe
__global__ __launch_bounds__(CDIM)
void k_gather(const float* __restrict__ wpe, const int* __restrict__ sel,
              float* __restrict__ X) {
    int slot = blockIdx.x;           // b*NP + i
    int b = slot >> 3;
    int si = sel[slot];
    if (si < 0) si = 0;
    X[(size_t)slot * CDIM + threadIdx.x] =
        wpe[((size_t)b * LW + si) * CDIM + threadIdx.x];
}

// folded bias: out[n] = sum_k bvec[k]*Bm[k,n] + badd[n]
__global__ __launch_bounds__(CDIM)
void k_fold_bias(const float* __restrict__ bvec, const float* __restrict__ Bm, int ldb,
                 const float* __restrict__ badd, float* __restrict__ outb) {
    int n = threadIdx.x;
    float s = badd[n];
    for (int k = 0; k < CDIM; k++) s += bvec[k] * Bm[(size_t)k * ldb + n];
    outb[n] = s;
}

// =======================================================================
// scores + softmax per (b,h): attn[q,k] = softmax_k(qp.kp/sqrt(Dh)) w/ mask
// =======================================================================
__global__ __launch_bounds__(256)
void k_attn_scores(const float* __restrict__ QP, const float* __restrict__ KP,
                   const unsigned char* __restrict__ tmask, float* __restrict__ AH) {
    __shared__ float sQ[NP * DH];
    __shared__ float sS[NP][LW];
    __shared__ float smx[NP], ssum[NP];
    int bh = blockIdx.x;
    int b = bh >> 3, h = bh & 7;
    int t = threadIdx.x;
    {
        int q = t >> 5, d = t & 31;
        sQ[t] = QP[((size_t)(b * NP + q)) * CDIM + h * DH + d];
    }
    __syncthreads();
    const float* krow = KP + ((size_t)(b * LW + t)) * CDIM + h * DH;
    float kv[DH];
#pragma unroll
    for (int d = 0; d < DH; d++) kv[d] = krow[d];
    bool km = tmask[(size_t)b * LTXT + 1 + t] != 0;
    const float scale = 0.17677669529663687f;   // 1/sqrt(32)
#pragma unroll
    for (int q = 0; q < NP; q++) {
        float s = 0.0f;
#pragma unroll
        for (int d = 0; d < DH; d++) s += sQ[q * DH + d] * kv[d];
        sS[q][t] = km ? s * scale : NEG_INF;
    }
    __syncthreads();
    if (t < NP) {
        float mx = NEG_INF;
        for (int k = 0; k < LW; k++) mx = fmaxf(mx, sS[t][k]);
        float su = 0.0f;
        for (int k = 0; k < LW; k++) {
            float v = sS[t][k];
            su += (v == NEG_INF) ? 0.0f : expf(v - mx);
        }
        smx[t] = mx; ssum[t] = su;
    }
    __syncthreads();
#pragma unroll
    for (int q = 0; q < NP; q++) {
        float v = sS[q][t];
        float a = (v == NEG_INF) ? 0.0f : expf(v - smx[q]) / ssum[q];
        AH[(((size_t)bh) * NP + q) * LW + t] = a;
    }
}

// context: ctx[b,q,h*32+d] = sum_k attn[b,h,q,k] * vp2[b,k,h*32+d]
__global__ __launch_bounds__(256)
void k_attn_ctx(const float* __restrict__ AH, const float* __restrict__ VP2,
                float* __restrict__ CTX) {
    int bh = blockIdx.x;
    int b = bh >> 3, h = bh & 7;
    int t = threadIdx.x;
    int q = t >> 5, d = t & 31;
    const float* arow = AH + ((size_t)bh * NP + q) * LW;
    const float* vcol = VP2 + (size_t)(b * LW) * CDIM + h * DH + d;
    float acc = 0.0f;
    for (int k = 0; k < LW; k++) acc += arow[k] * vcol[(size_t)k * CDIM];
    CTX[((size_t)(b * NP + q)) * CDIM + h * DH + d] = acc;
}

// slot_sim = mean over heads of final-layer attention
__global__ __launch_bounds__(256)
void k_slot_sim(const float* __restrict__ AH, float* __restrict__ out) {
    int bq = blockIdx.x;             // b*NP + q
    int b = bq >> 3, q = bq & 7;
    int k = threadIdx.x;
    float s = 0.0f;
#pragma unroll
    for (int h = 0; h < NH; h++)
        s += AH[(((size_t)(b * NH + h)) * NP + q) * LW + k];
    out[(size_t)bq * LW + k] = s * 0.125f;
}

// X = LayerNorm(X + R) * g + b  (in place on X)
__global__ __launch_bounds__(CDIM)
void k_ln_add(float* __restrict__ X, const float* __restrict__ R,
              const float* __restrict__ g, const float* __restrict__ bt) {
    __shared__ float red[CDIM];
    int row = blockIdx.x, c = threadIdx.x;
    float v = X[(size_t)row * CDIM + c] + R[(size_t)row * CDIM + c];
    red[c] = v; __syncthreads();
    for (int off = 128; off > 0; off >>= 1) {
        if (c < off) red[c] += red[c + off];
        __syncthreads();
    }
    float mu = red[0] * (1.0f / CDIM); __syncthreads();
    float dv = v - mu;
    red[c] = dv * dv; __syncthreads();
    for (int off = 128; off > 0; off >>= 1) {
        if (c < off) red[c] += red[c + off];
        __syncthreads();
    }
    float var = red[0] * (1.0f / CDIM);
    X[(size_t)row * CDIM + c] = dv / sqrtf(var + 1e-5f) * g[c] + bt[c];
}

__global__ __launch_bounds__(CDIM)
void k_copy(const float* __restrict__ src, float* __restrict__ dst) {
    size_t i = (size_t)blockIdx.x * CDIM + threadIdx.x;
    dst[i] = src[i];
}

__global__ __launch_bounds__(CDIM)
void k_eos(const float* __restrict__ eos, float* __restrict__ out) {
    out[(size_t)blockIdx.x * CDIM + threadIdx.x] = eos[threadIdx.x];
}

// =======================================================================
extern "C" void kernel_launch(void* const* d_in, const int* in_sizes, int n_in,
                              void* d_out, int out_size, void* d_ws, size_t ws_size,
                              hipStream_t stream) {
    (void)in_sizes; (void)n_in; (void)out_size; (void)ws_size;
    const float* txt_emb     = (const float*)d_in[0];
    const unsigned char* txt_mask   = (const unsigned char*)d_in[1];
    const float* video_feats = (const float*)d_in[2];
    const unsigned char* video_mask = (const unsigned char*)d_in[3];
    const float* wproj_w = (const float*)d_in[4];
    const float* wproj_b = (const float*)d_in[5];
    const float* vproj_w = (const float*)d_in[6];
    const float* vproj_b = (const float*)d_in[7];
    const float* q_w   = (const float*)d_in[8];
    const float* q_b   = (const float*)d_in[9];
    const float* kv_w  = (const float*)d_in[10];
    const float* kv_b  = (const float*)d_in[11];
    const float* in_w  = (const float*)d_in[12];
    const float* in_b  = (const float*)d_in[13];
    const float* out_w = (const float*)d_in[14];
    const float* out_b = (const float*)d_in[15];
    const float* ln0_g = (const float*)d_in[16];
    const float* ln0_b = (const float*)d_in[17];
    const float* lin_w = (const float*)d_in[18];
    const float* lin_b = (const float*)d_in[19];
    const float* ln1_g = (const float*)d_in[20];
    const float* ln1_b = (const float*)d_in[21];
    const float* eos_slot = (const float*)d_in[22];

    float* ws  = (float*)d_ws;
    float* out = (float*)d_out;

    // ---- positional encoding + word_pe ----
    k_scan_mask<<<BATCH, LW, 0, stream>>>(txt_mask, ws + CUM_OFF);
    k_build_wpe<<<BATCH * LW, CDIM, 0, stream>>>(txt_emb, ws + CUM_OFF, ws + WPE_OFF);

    // ---- wp / vp projections (LDS-tiled WMMA fp32) ----
    k_gemm64<<<(BATCH * LW / 64) * (CDIM / 64), 256, 0, stream>>>(
        ws + WPE_OFF, CDIM, wproj_w, CDIM, wproj_b, ws + WP_OFF, CDIM,
        CDIM / 64, CDIM, 0);
    k_gemm64<<<(BATCH * TV / 64) * (CDIM / 64), 256, 0, stream>>>(
        video_feats, CDIM, vproj_w, CDIM, vproj_b, ws + VPV_OFF, CDIM,
        CDIM / 64, CDIM, 0);

    // ---- sim = wp @ vp^T directly into d_out attn region, softmax in place ----
    k_gemm64_sim<<<BATCH * (LW / 64) * (TV / 64), 256, 0, stream>>>(
        ws + WP_OFF, ws + VPV_OFF, out + OUT_AW);
    k_softmax_entropy<<<BATCH * LW, 256, 0, stream>>>(out + OUT_AW, video_mask,
                                                      ws + ENT_OFF);

    // ---- selection + gather ----
    k_select<<<BATCH, 256, 0, stream>>>(ws + ENT_OFF, txt_mask, (int*)(ws + SEL_OFF));
    k_gather<<<BATCH * NP, CDIM, 0, stream>>>(ws + WPE_OFF, (const int*)(ws + SEL_OFF),
                                              ws + X_OFF);

    // ---- 3 cross-attention layers ----
    for (int l = 0; l < NL; l++) {
        const float* qw   = q_w   + (size_t)l * CDIM * CDIM;
        const float* qb   = q_b   + (size_t)l * CDIM;
        const float* kvw  = kv_w  + (size_t)l * CDIM * 2 * CDIM;
        const float* kvb  = kv_b  + (size_t)l * 2 * CDIM;
        const float* inw  = in_w  + (size_t)l * CDIM * 3 * CDIM;
        const float* inb  = in_b  + (size_t)l * 3 * CDIM;
        const float* ow   = out_w + (size_t)l * CDIM * CDIM;
        const float* ob   = out_b + (size_t)l * CDIM;
        const float* lw   = lin_w + (size_t)l * CDIM * CDIM;
        const float* lb   = lin_b + (size_t)l * CDIM;
        const float* g0 = ln0_g + (size_t)l * CDIM, *b0 = ln0_b + (size_t)l * CDIM;
        const float* g1 = ln1_g + (size_t)l * CDIM, *b1 = ln1_b + (size_t)l * CDIM;

        // fold back-to-back linear layers: Wq_f = qw @ in_w[:, :C], etc. (256x256)
        k_gemm64<<<16, 256, 0, stream>>>(qw, CDIM, inw, 3 * CDIM, nullptr,
                                         ws + WQF_OFF, CDIM, CDIM / 64, CDIM, 0);
        k_gemm64<<<16, 256, 0, stream>>>(kvw, 2 * CDIM, inw + CDIM, 3 * CDIM, nullptr,
                                         ws + WKF_OFF, CDIM, CDIM / 64, CDIM, 0);
        k_gemm64<<<16, 256, 0, stream>>>(kvw + CDIM, 2 * CDIM, inw + 2 * CDIM, 3 * CDIM,
                                         nullptr, ws + WVF_OFF, CDIM, CDIM / 64, CDIM, 0);
        k_fold_bias<<<1, CDIM, 0, stream>>>(qb, inw, 3 * CDIM, inb, ws + BQF_OFF);
        k_fold_bias<<<1, CDIM, 0, stream>>>(kvb, inw + CDIM, 3 * CDIM, inb + CDIM,
                                            ws + BKF_OFF);
        k_fold_bias<<<1, CDIM, 0, stream>>>(kvb + CDIM, inw + 2 * CDIM, 3 * CDIM,
                                            inb + 2 * CDIM, ws + BVF_OFF);

        // qp = X @ Wq_f + bq_f (1024 x 256)
        k_gemm64<<<(BATCH * NP / 64) * (CDIM / 64), 256, 0, stream>>>(
            ws + X_OFF, CDIM, ws + WQF_OFF, CDIM, ws + BQF_OFF, ws + QP_OFF, CDIM,
            CDIM / 64, CDIM, 0);
        // kp = word_pe @ Wk_f + bk_f (32768 x 256), reuse WP region
        k_gemm64<<<(BATCH * LW / 64) * (CDIM / 64), 256, 0, stream>>>(
            ws + WPE_OFF, CDIM, ws + WKF_OFF, CDIM, ws + BKF_OFF, ws + WP_OFF, CDIM,
            CDIM / 64, CDIM, 0);
        // vp2 = word_pe @ Wv_f + bv_f, reuse VPV region
        k_gemm64<<<(BATCH * LW / 64) * (CDIM / 64), 256, 0, stream>>>(
            ws + WPE_OFF, CDIM, ws + WVF_OFF, CDIM, ws + BVF_OFF, ws + VPV_OFF, CDIM,
            CDIM / 64, CDIM, 0);

        k_attn_scores<<<BATCH * NH, 256, 0, stream>>>(ws + QP_OFF, ws + WP_OFF,
                                                      txt_mask, ws + AH_OFF);
        k_attn_ctx<<<BATCH * NH, 256, 0, stream>>>(ws + AH_OFF, ws + VPV_OFF,
                                                   ws + CTX_OFF);
        // attn_out = ctx @ out_w + out_b
        k_gemm64<<<(BATCH * NP / 64) * (CDIM / 64), 256, 0, stream>>>(
            ws + CTX_OFF, CDIM, ow, CDIM, ob, ws + AO_OFF, CDIM, CDIM / 64, CDIM, 0);
        k_ln_add<<<BATCH * NP, CDIM, 0, stream>>>(ws + X_OFF, ws + AO_OFF, g0, b0);
        // update = relu(X @ lin_w + lin_b)
        k_gemm64<<<(BATCH * NP / 64) * (CDIM / 64), 256, 0, stream>>>(
            ws + X_OFF, CDIM, lw, CDIM, lb, ws + AO_OFF, CDIM, CDIM / 64, CDIM, 1);
        k_ln_add<<<BATCH * NP, CDIM, 0, stream>>>(ws + X_OFF, ws + AO_OFF, g1, b1);

        if (l == NL - 1)
            k_slot_sim<<<BATCH * NP, 256, 0, stream>>>(ws + AH_OFF, out + OUT_SS);
    }

    // ---- final outputs ----
    k_copy<<<BATCH * NP, CDIM, 0, stream>>>(ws + X_OFF, out + OUT_PS);
    k_eos<<<BATCH, CDIM, 0, stream>>>(eos_slot, out + OUT_EOS);
}